// inference_model_86990267613480
// MI455X (gfx1250) — compile-verified
//
#include <hip/hip_runtime.h>
#include <hip/hip_bf16.h>
#include <math.h>

// ---------------------------------------------------------------------------
// Sizes (fixed by the reference): B=8, EVI=5, BT=40, L=128, D=768, K=21
// ---------------------------------------------------------------------------
#define BT   40
#define LL   128
#define DD   768
#define KK   21
#define NPAIR 200   // 8 * 5 * 5

typedef __attribute__((ext_vector_type(2))) float v2f;
typedef __attribute__((ext_vector_type(8))) float v8f;

// Gaussian kernel constants: mu[0]=1.0, mu[k]=0.95-(k-1)*0.1 ; sigma[0]=1e-3, rest 0.1
__constant__ float c_mu[KK] = {
  1.00f, 0.95f, 0.85f, 0.75f, 0.65f, 0.55f, 0.45f, 0.35f, 0.25f, 0.15f, 0.05f,
 -0.05f,-0.15f,-0.25f,-0.35f,-0.45f,-0.55f,-0.65f,-0.75f,-0.85f,-0.95f };
__constant__ float c_i2s2[KK] = {  // 1/(2*sigma^2)
  500000.0f, 50.f,50.f,50.f,50.f,50.f,50.f,50.f,50.f,50.f,50.f,
  50.f,50.f,50.f,50.f,50.f,50.f,50.f,50.f,50.f,50.f };

// ---------------------------------------------------------------------------
// K1: h[bt] = tpool[bt]^T @ hiddens[bt]    (M=128 rows m, N=768 cols d, K=128 l)
// f32 WMMA 16x16x4. One wave per 16x16 output tile.
// grid (6, 8, 40), block 256 (8 waves cover 8 d-subtiles).
// ---------------------------------------------------------------------------
__global__ void k1_h_gemm(const float* __restrict__ hiddens,
                          const float* __restrict__ tpool,
                          float* __restrict__ h)
{
  const int bt    = blockIdx.z;
  const int mtile = blockIdx.y * 16;
  const int wave  = threadIdx.x >> 5;
  const int lane  = threadIdx.x & 31;
  const int dtile = (blockIdx.x * 8 + wave) * 16;

  const float* Hd = hiddens + (size_t)bt * LL * DD;   // [l][d]
  const float* Tp = tpool   + (size_t)bt * LL * LL;   // [l][m]
  const int col  = lane & 15;
  const int koff = (lane >> 4) * 2;

  v8f c = {0.f,0.f,0.f,0.f,0.f,0.f,0.f,0.f};
  for (int k0 = 0; k0 < LL; k0 += 4) {
    const int k = k0 + koff;
    v2f a, b;
    // A[m][l] = tpool[l][m]  (strided loads)
    a.x = Tp[(size_t)(k + 0) * LL + (mtile + col)];
    a.y = Tp[(size_t)(k + 1) * LL + (mtile + col)];
    // B[l][d] = hiddens[l][d]
    b.x = Hd[(size_t)(k + 0) * DD + (dtile + col)];
    b.y = Hd[(size_t)(k + 1) * DD + (dtile + col)];
    c = __builtin_amdgcn_wmma_f32_16x16x4_f32(false, a, false, b,
                                              (short)0, c, false, false);
  }
  float* Ho = h + (size_t)bt * LL * DD;
  const int mbase = mtile + (lane >> 4) * 8;
  #pragma unroll
  for (int r = 0; r < 8; ++r)
    Ho[(size_t)(mbase + r) * DD + dtile + col] = c[r];
}

// ---------------------------------------------------------------------------
// K2: hn = l2norm(h) along d.  grid (40*128), block 256.
// ---------------------------------------------------------------------------
__global__ void k2_l2norm(const float* __restrict__ h, float* __restrict__ hn)
{
  const int row = blockIdx.x;
  const float* src = h  + (size_t)row * DD;
  float*       dst = hn + (size_t)row * DD;
  __shared__ float red[256];
  float ss = 0.f;
  for (int d = threadIdx.x; d < DD; d += 256) { float v = src[d]; ss += v * v; }
  red[threadIdx.x] = ss; __syncthreads();
  for (int s = 128; s > 0; s >>= 1) {
    if (threadIdx.x < s) red[threadIdx.x] += red[threadIdx.x + s];
    __syncthreads();
  }
  const float inv = 1.f / fmaxf(sqrtf(red[0]), 1e-12f);
  for (int d = threadIdx.x; d < DD; d += 256) dst[d] = src[d] * inv;
}

// ---------------------------------------------------------------------------
// K3: all pairwise similarity matrices
//   S[p=b*25+e*5+i][l][m] = sum_d hn[b*5+e][l][d] * hn[b*5+i][m][d]
// f32 WMMA 16x16x4, K=768. grid (8 ltiles, 200 pairs), block 256 (8 m-tiles).
// Both fragments read 2 consecutive d's per lane -> 8B vector loads.
// ---------------------------------------------------------------------------
__global__ void k3_sim_gemm(const float* __restrict__ hn, float* __restrict__ S)
{
  const int p = blockIdx.y;
  const int b = p / 25;
  const int e = (p % 25) / 5;
  const int i = p % 5;
  const int ltile = blockIdx.x * 16;
  const int wave  = threadIdx.x >> 5;
  const int lane  = threadIdx.x & 31;
  const int mtile = wave * 16;
  const int col   = lane & 15;
  const int koff  = (lane >> 4) * 2;

  const float* Arow = hn + (size_t)(b * 5 + e) * LL * DD + (size_t)(ltile + col) * DD;
  const float* Brow = hn + (size_t)(b * 5 + i) * LL * DD + (size_t)(mtile + col) * DD;

  v8f c = {0.f,0.f,0.f,0.f,0.f,0.f,0.f,0.f};
  for (int k0 = 0; k0 < DD; k0 += 4) {
    const int k = k0 + koff;                       // even -> 8B aligned
    const v2f a = *reinterpret_cast<const v2f*>(Arow + k);
    const v2f bb = *reinterpret_cast<const v2f*>(Brow + k);
    c = __builtin_amdgcn_wmma_f32_16x16x4_f32(false, a, false, bb,
                                              (short)0, c, false, false);
  }
  float* So = S + (size_t)p * LL * LL;
  const int lbase = ltile + (lane >> 4) * 8;
  #pragma unroll
  for (int r = 0; r < 8; ++r)
    So[(size_t)(lbase + r) * LL + mtile + col] = c[r];
}

// ---------------------------------------------------------------------------
// K4: claim-side kernel pooling -> sel[bt].  grid 40, block 128 (thread = l).
// Uses diagonal pair S[b,e,e].  Linearity: fold log(pool)*W_sel per thread.
// ---------------------------------------------------------------------------
__global__ void k4_sel(const float* __restrict__ S,
                       const int* __restrict__ comb_msk,
                       const int* __restrict__ comb_seg,
                       const float* __restrict__ W_sel,
                       const float* __restrict__ b_sel,
                       float* __restrict__ sel)
{
  const int bt = blockIdx.x;
  const int b  = bt / 5, e = bt % 5;
  const int p  = b * 25 + e * 5 + e;
  const int l  = threadIdx.x;

  __shared__ float sm_ev[LL];
  __shared__ float red [LL];
  __shared__ float red2[LL];

  const float mt_l  = (l == 0) ? 0.f : (float)comb_msk[bt * LL + l];
  const float seg_l = (float)comb_seg[bt * LL + l];
  const float mc_l  = (1.f - seg_l) * mt_l;        // mask_claim
  sm_ev[l] = seg_l * mt_l;                         // mask_ev
  __syncthreads();

  const float* Srow = S + (size_t)p * LL * LL + (size_t)l * LL;
  float pool[KK];
  #pragma unroll
  for (int k = 0; k < KK; ++k) pool[k] = 0.f;
  for (int m = 0; m < LL; ++m) {
    if (sm_ev[m] > 0.f) {
      const float s = Srow[m];
      #pragma unroll
      for (int k = 0; k < KK; ++k) {
        const float d = s - c_mu[k];
        pool[k] += expf(-d * d * c_i2s2[k]);
      }
    }
  }
  float acc = 0.f;
  #pragma unroll
  for (int k = 0; k < KK; ++k)
    acc += logf(fmaxf(pool[k], 1e-10f)) * W_sel[k];
  acc *= mc_l;

  red[l] = acc; red2[l] = mc_l; __syncthreads();
  for (int s = 64; s > 0; s >>= 1) {
    if (l < s) { red[l] += red[l + s]; red2[l] += red2[l + s]; }
    __syncthreads();
  }
  if (l == 0) sel[bt] = red[0] / (red2[0] + 1e-10f) + b_sel[0];
}

// ---------------------------------------------------------------------------
// K5: evidence attention + denoise.  grid 200 (q = i*40+b*5+e), block 128.
//   pool over m masked by mask_text[b*5+i]; logit = lp.W_att + b_att;
//   masked softmax over l; denoise[q][d] = sum_l att[l]*h[b*5+e][l][d].
// ---------------------------------------------------------------------------
__global__ void k5_att_denoise(const float* __restrict__ S,
                               const float* __restrict__ h,
                               const int* __restrict__ comb_msk,
                               const float* __restrict__ W_att,
                               const float* __restrict__ b_att,
                               float* __restrict__ den)
{
  const int q = blockIdx.x;
  const int i = q / 40;
  const int r = q % 40;
  const int b = r / 5, e = r % 5;
  const int p = b * 25 + e * 5 + i;
  const int l = threadIdx.x;
  const int bt_i = b * 5 + i, bt_e = b * 5 + e;

  __shared__ float own[LL];
  __shared__ float att[LL];
  __shared__ float red[LL];

  own[l] = (l == 0) ? 0.f : (float)comb_msk[bt_i * LL + l];
  const float mt_l = (l == 0) ? 0.f : (float)comb_msk[bt_e * LL + l];
  __syncthreads();

  const float* Srow = S + (size_t)p * LL * LL + (size_t)l * LL;
  float pool[KK];
  #pragma unroll
  for (int k = 0; k < KK; ++k) pool[k] = 0.f;
  for (int m = 0; m < LL; ++m) {
    if (own[m] > 0.f) {
      const float s = Srow[m];
      #pragma unroll
      for (int k = 0; k < KK; ++k) {
        const float d = s - c_mu[k];
        pool[k] += expf(-d * d * c_i2s2[k]);
      }
    }
  }
  float logit = b_att[0];
  #pragma unroll
  for (int k = 0; k < KK; ++k)
    logit += logf(fmaxf(pool[k], 1e-10f)) * W_att[k];
  if (!(mt_l > 0.f)) logit = -10000.f;

  // softmax over l
  red[l] = logit; __syncthreads();
  for (int s = 64; s > 0; s >>= 1) {
    if (l < s) red[l] = fmaxf(red[l], red[l + s]);
    __syncthreads();
  }
  const float mx = red[0]; __syncthreads();
  const float ex = expf(logit - mx);
  red[l] = ex; __syncthreads();
  for (int s = 64; s > 0; s >>= 1) {
    if (l < s) red[l] += red[l + s];
    __syncthreads();
  }
  att[l] = ex / red[0];
  __syncthreads();

  const float* Hrow = h + (size_t)bt_e * LL * DD;
  float* out = den + (size_t)q * DD;
  for (int d = l; d < DD; d += LL) {
    float acc = 0.f;
    for (int ll = 0; ll < LL; ++ll) acc += att[ll] * Hrow[(size_t)ll * DD + d];
    out[d] = acc;
  }
}

// ---------------------------------------------------------------------------
// K6: gate MLP + softmax over evidence -> att_de[b][i].  grid 40 (i*8+b), 128 thr.
// ---------------------------------------------------------------------------
__global__ void k6_gat(const float* __restrict__ pooled,
                       const float* __restrict__ den,
                       const float* __restrict__ W_gat1,
                       const float* __restrict__ b_gat1,
                       const float* __restrict__ W_gat2,
                       const float* __restrict__ b_gat2,
                       float* __restrict__ attde)
{
  const int i = blockIdx.x / 8;
  const int b = blockIdx.x % 8;
  const int tid = threadIdx.x;           // 0..127 = hidden unit j
  __shared__ float red[LL];
  __shared__ float g[5];
  __shared__ float w[5];

  const float* own = pooled + (size_t)(b * 5 + i) * DD;
  for (int e = 0; e < 5; ++e) {
    const float* dn   = den + (size_t)(i * 40 + b * 5 + e) * DD;
    const float* Wrow = W_gat1 + (size_t)tid * (2 * DD);
    float acc = b_gat1[tid];
    for (int d = 0; d < DD; ++d) acc += Wrow[d] * own[d];
    for (int d = 0; d < DD; ++d) acc += Wrow[DD + d] * dn[d];
    acc = fmaxf(acc, 0.f);               // relu
    red[tid] = acc * W_gat2[tid];
    __syncthreads();
    for (int s = 64; s > 0; s >>= 1) {
      if (tid < s) red[tid] += red[tid + s];
      __syncthreads();
    }
    if (tid == 0) g[e] = red[0] + b_gat2[0];
    __syncthreads();
  }
  if (tid == 0) {
    float mx = g[0];
    for (int e = 1; e < 5; ++e) mx = fmaxf(mx, g[e]);
    float s = 0.f;
    for (int e = 0; e < 5; ++e) { w[e] = expf(g[e] - mx); s += w[e]; }
    for (int e = 0; e < 5; ++e) w[e] /= s;
  }
  __syncthreads();
  float* out = attde + (size_t)(b * 5 + i) * DD;
  for (int d = tid; d < DD; d += LL) {
    float acc = 0.f;
    for (int e = 0; e < 5; ++e)
      acc += w[e] * den[(size_t)(i * 40 + b * 5 + e) * DD + d];
    out[d] = acc;
  }
}

// ---------------------------------------------------------------------------
// K7: classifier head + select/class softmax + log-mix.  grid 8, block 256.
// ---------------------------------------------------------------------------
__global__ void k7_final(const float* __restrict__ pooled,
                         const float* __restrict__ attde,
                         const float* __restrict__ sel,
                         const float* __restrict__ W_inf,
                         const float* __restrict__ b_inf,
                         float* __restrict__ out)
{
  const int b = blockIdx.x;
  const int tid = threadIdx.x;
  __shared__ float red[256];
  __shared__ float feat[5][3];

  for (int e = 0; e < 5; ++e) {
    const float* po = pooled + (size_t)(b * 5 + e) * DD;
    const float* ad = attde  + (size_t)(b * 5 + e) * DD;
    for (int c = 0; c < 3; ++c) {
      const float* Wr = W_inf + (size_t)c * (2 * DD);
      float acc = 0.f;
      for (int d = tid; d < DD; d += 256)
        acc += Wr[d] * po[d] + Wr[DD + d] * ad[d];
      red[tid] = acc; __syncthreads();
      for (int s = 128; s > 0; s >>= 1) {
        if (tid < s) red[tid] += red[tid + s];
        __syncthreads();
      }
      if (tid == 0) feat[e][c] = red[0] + b_inf[c];
      __syncthreads();
    }
  }
  if (tid == 0) {
    // select softmax over evidence
    float smax = sel[b * 5];
    for (int e = 1; e < 5; ++e) smax = fmaxf(smax, sel[b * 5 + e]);
    float sp[5], ssum = 0.f;
    for (int e = 0; e < 5; ++e) { sp[e] = expf(sel[b * 5 + e] - smax); ssum += sp[e]; }
    float accc[3] = {0.f, 0.f, 0.f};
    for (int e = 0; e < 5; ++e) {
      float cmx = fmaxf(feat[e][0], fmaxf(feat[e][1], feat[e][2]));
      float c0 = expf(feat[e][0] - cmx), c1 = expf(feat[e][1] - cmx), c2 = expf(feat[e][2] - cmx);
      float cs = c0 + c1 + c2;
      float w = sp[e] / ssum;
      accc[0] += w * (c0 / cs);
      accc[1] += w * (c1 / cs);
      accc[2] += w * (c2 / cs);
    }
    for (int c = 0; c < 3; ++c) out[b * 3 + c] = logf(accc[c]);
  }
}

// ---------------------------------------------------------------------------
// Launcher.  Workspace layout (floats), total ~45.3 MB:
//   h[40*128*768] | hn[40*128*768] | S[200*128*128] | sel[64] |
//   den[200*768]  | attde[40*768]
// ---------------------------------------------------------------------------
extern "C" void kernel_launch(void* const* d_in, const int* in_sizes, int n_in,
                              void* d_out, int out_size, void* d_ws, size_t ws_size,
                              hipStream_t stream)
{
  const float* hiddens  = (const float*)d_in[0];
  const float* pooled   = (const float*)d_in[1];
  const float* tpool    = (const float*)d_in[2];
  const int*   comb_msk = (const int*)  d_in[3];
  const int*   comb_seg = (const int*)  d_in[4];
  const float* W_att    = (const float*)d_in[5];
  const float* b_att    = (const float*)d_in[6];
  const float* W_sel    = (const float*)d_in[7];
  const float* b_sel    = (const float*)d_in[8];
  const float* W_gat1   = (const float*)d_in[9];
  const float* b_gat1   = (const float*)d_in[10];
  const float* W_gat2   = (const float*)d_in[11];
  const float* b_gat2   = (const float*)d_in[12];
  const float* W_inf    = (const float*)d_in[13];
  const float* b_inf    = (const float*)d_in[14];

  float* ws    = (float*)d_ws;
  float* h     = ws;
  float* hn    = h   + (size_t)BT * LL * DD;
  float* S     = hn  + (size_t)BT * LL * DD;
  float* sel   = S   + (size_t)NPAIR * LL * LL;
  float* den   = sel + 64;
  float* attde = den + (size_t)NPAIR * DD;
  float* out   = (float*)d_out;

  k1_h_gemm     <<<dim3(6, 8, BT), 256, 0, stream>>>(hiddens, tpool, h);
  k2_l2norm     <<<dim3(BT * LL),  256, 0, stream>>>(h, hn);
  k3_sim_gemm   <<<dim3(8, NPAIR), 256, 0, stream>>>(hn, S);
  k4_sel        <<<dim3(BT),       LL,  0, stream>>>(S, comb_msk, comb_seg, W_sel, b_sel, sel);
  k5_att_denoise<<<dim3(NPAIR),    LL,  0, stream>>>(S, h, comb_msk, W_att, b_att, den);
  k6_gat        <<<dim3(BT),       LL,  0, stream>>>(pooled, den, W_gat1, b_gat1, W_gat2, b_gat2, attde);
  k7_final      <<<dim3(8),        256, 0, stream>>>(pooled, attde, sel, W_inf, b_inf, out);
}